// DilatedKnn2d_42339787604138
// MI455X (gfx1250) — compile-verified
//
#include <hip/hip_runtime.h>

typedef __attribute__((ext_vector_type(2))) float v2f;
typedef __attribute__((ext_vector_type(8))) float v8f;

#define BN   4
#define NP   8192
#define NC   64
#define KBIG 32
#define KOUT 16
#define TILE 16
#define NEG_INF (-3.402823466e38f)

// One wave (32 lanes) per 16-row tile. Fused f32-WMMA Gram matrix + streaming top-32.
__global__ __launch_bounds__(32) void dilated_knn_kernel(const float* __restrict__ x,
                                                         int* __restrict__ out) {
    __shared__ float sTile[TILE * TILE];     // 16x16 score tile
    __shared__ float sVal[KBIG * 32];        // per-lane top-32 values  [entry][lane]
    __shared__ int   sIdx[KBIG * 32];        // per-lane top-32 indices [entry][lane]

    const int lane = threadIdx.x;                // 0..31
    const int b    = blockIdx.y;
    const int n0   = blockIdx.x * TILE;
    const float* __restrict__ Xb = x + (size_t)b * NC * NP;   // x[b][c][n] : c stride NP

    const int mrow = lane & 15;                  // M (A) / N (B) position for this lane
    const int koff = (lane < 16) ? 0 : 2;        // ISA A/B K-striping across lane halves

    // ---- init top-k lists ----
    for (int e = 0; e < KBIG; ++e) {
        sVal[e * 32 + lane] = NEG_INF;
        sIdx[e * 32 + lane] = 0;
    }
    float minval = NEG_INF;
    int   minpos = 0;

    // ---- load A tile (16 rows x 64 channels) as 16 K-slices of 16x4 ----
    v2f A[16];
#pragma unroll
    for (int kk = 0; kk < 16; ++kk) {
        const float* p = Xb + (size_t)(kk * 4 + koff) * NP + (n0 + mrow);
        A[kk].x = p[0];          // K = kk*4 + koff
        A[kk].y = p[NP];         // K = kk*4 + koff + 1
    }

    const int colbase = (lane >> 4) * 8;         // lane owns row mrow, cols colbase..+7

    // ---- stream over all 8192 columns in 16-wide tiles ----
    for (int m0 = 0; m0 < NP; m0 += TILE) {
        // load B tile (64 x 16 slice of X^T) + partial column norms
        v2f  Bt[16];
        float pn = 0.f;
#pragma unroll
        for (int kk = 0; kk < 16; ++kk) {
            const float* p = Xb + (size_t)(kk * 4 + koff) * NP + (m0 + mrow);
            float bx = p[0], by = p[NP];
            Bt[kk].x = bx; Bt[kk].y = by;
            pn += bx * bx + by * by;
        }
        // full ||x_m||^2 for column m0+mrow: combine the two lane-half partial sums
        float colnorm = pn + __shfl_xor(pn, 16);

        // 16 chained f32 WMMAs: D(16x16) = A(16x64) * B(64x16)
        v8f acc = {0.f, 0.f, 0.f, 0.f, 0.f, 0.f, 0.f, 0.f};
#pragma unroll
        for (int kk = 0; kk < 16; ++kk) {
            acc = __builtin_amdgcn_wmma_f32_16x16x4_f32(
                false, A[kk], false, Bt[kk], (short)0, acc, false, false);
        }

        __syncthreads();  // WAR on sTile vs previous iteration's reads
        // D layout: VGPR r -> M=r (lanes 0-15) / M=r+8 (lanes 16-31), N=mrow.
        // Store ranking score 2*dot - ||x_m||^2 directly.
        const int rbase = (lane < 16) ? 0 : 8;
#pragma unroll
        for (int r = 0; r < 8; ++r) {
            sTile[(rbase + r) * TILE + mrow] = 2.f * acc[r] - colnorm;
        }
        __syncthreads();

        // streaming top-32: each lane scans its 8 columns of its row
#pragma unroll
        for (int j = 0; j < 8; ++j) {
            float v = sTile[mrow * TILE + colbase + j];
            if (v > minval) {                     // ties keep earlier (smaller) index
                sVal[minpos * 32 + lane] = v;
                sIdx[minpos * 32 + lane] = m0 + colbase + j;
                float mv = sVal[lane];            // rescan for new min
                int   mp = 0;
                for (int e = 1; e < KBIG; ++e) {
                    float w = sVal[e * 32 + lane];
                    if (w < mv) { mv = w; mp = e; }
                }
                minval = mv; minpos = mp;
            }
        }
    }
    __syncthreads();

    // ---- merge the two half-row lists, select top-32 (value desc, index asc),
    //      emit every DILATION-th (==2) rank ----
    if (lane < 16) {
        const int n = n0 + lane;
        int* nnOut = out + (((size_t)0 * BN + b) * NP + n) * KOUT;
        int* ctOut = out + (((size_t)1 * BN + b) * NP + n) * KOUT;

        for (int rank = 0; rank < KBIG; ++rank) {
            float bestv = NEG_INF;
            int   besti = 0x7fffffff;
            int   bests = -1;
            for (int e = 0; e < 2 * KBIG; ++e) {
                const int l    = (e < KBIG) ? lane : (lane + 16);
                const int slot = e & (KBIG - 1);
                float w  = sVal[slot * 32 + l];
                int   wi = sIdx[slot * 32 + l];
                if (w > bestv || (w == bestv && wi < besti)) {
                    bestv = w; besti = wi; bests = e;
                }
            }
            const int l = (bests < KBIG) ? lane : (lane + 16);
            sVal[(bests & (KBIG - 1)) * 32 + l] = NEG_INF;   // consume
            if ((rank & 1) == 0) nnOut[rank >> 1] = besti;
        }
        for (int j = 0; j < KOUT; ++j) ctOut[j] = n;
    }
}

extern "C" void kernel_launch(void* const* d_in, const int* in_sizes, int n_in,
                              void* d_out, int out_size, void* d_ws, size_t ws_size,
                              hipStream_t stream) {
    (void)in_sizes; (void)n_in; (void)d_ws; (void)ws_size; (void)out_size;
    const float* xin = (const float*)d_in[0];
    int* out = (int*)d_out;
    dim3 grid(NP / TILE, BN);
    dim3 block(32);
    dilated_knn_kernel<<<grid, block, 0, stream>>>(xin, out);
}